// HealerGAT_32581621907479
// MI455X (gfx1250) — compile-verified
//
#include <hip/hip_runtime.h>
#include <math.h>

typedef __attribute__((ext_vector_type(2))) float v2f;
typedef __attribute__((ext_vector_type(8))) float v8f;

#define NEG_SLOPE 0.2f

// ---------------------------------------------------------------------------
// WMMA fp32 GEMM:  C[M x Fout] = A[M x K] @ W[K x Fout]
// One wave computes a 16 x Fout strip (NT = Fout/16 tiles), K consumed 4 at a
// time via V_WMMA_F32_16X16X4_F32.  A 16x4 f32 layout: lanes 0-15 hold K=0,1
// (VGPR0/1), lanes 16-31 hold K=2,3 -> a single 8-byte vector load per lane.
// ---------------------------------------------------------------------------
template <int NT>
__global__ void gat455_gemm_wmma(const float* __restrict__ A,
                                 const float* __restrict__ W,
                                 float* __restrict__ C,
                                 int M, int K, int Fout) {
  int wave = (blockIdx.x * blockDim.x + threadIdx.x) >> 5;
  int lane = threadIdx.x & 31;
  int nRowTiles = M >> 4;
  if (wave >= nRowTiles) return;              // uniform per wave -> EXEC all-1s

  const int row0  = wave << 4;
  const int arow  = row0 + (lane & 15);
  const int kHalf = (lane >> 4) << 1;         // 0 for lanes 0-15, 2 for 16-31
  const int nsel  = lane & 15;
  const float* __restrict__ aRow = A + (size_t)arow * K + kHalf;

  v8f acc[NT] = {};

  for (int k0 = 0; k0 < K; k0 += 4) {
    v2f a = *(const v2f*)(aRow + k0);          // global_load_b64
#pragma unroll
    for (int t = 0; t < NT; ++t) {
      int col = (t << 4) + nsel;
      v2f b;
      b.x = W[(size_t)(k0 + kHalf) * Fout + col];
      b.y = W[(size_t)(k0 + kHalf + 1) * Fout + col];
      acc[t] = __builtin_amdgcn_wmma_f32_16x16x4_f32(
          false, a, false, b, (short)0, acc[t], false, false);
    }
  }

  const int crow = row0 + ((lane >> 4) << 3);
#pragma unroll
  for (int t = 0; t < NT; ++t) {
#pragma unroll
    for (int v = 0; v < 8; ++v) {
      C[(size_t)(crow + v) * Fout + (t << 4) + nsel] = acc[t][v];
    }
  }
}

// ---------------------------------------------------------------------------
// Per-(node, head) attention coefficients (float4 dot products).
// ---------------------------------------------------------------------------
__global__ void gat455_alpha(const float* __restrict__ h,
                             const float* __restrict__ a_src,
                             const float* __restrict__ a_dst,
                             float* __restrict__ as, float* __restrict__ ad,
                             int nNodes, int H, int Cc) {
  int i = blockIdx.x * blockDim.x + threadIdx.x;
  if (i >= nNodes * H) return;
  int n = i / H, hh = i - n * H;
  const float4* hv  = (const float4*)(h + (size_t)n * H * Cc + hh * Cc);
  const float4* asv = (const float4*)(a_src + hh * Cc);
  const float4* adv = (const float4*)(a_dst + hh * Cc);
  float s = 0.f, d = 0.f;
  for (int c = 0; c < (Cc >> 2); ++c) {
    float4 v = hv[c], va = asv[c], vb = adv[c];
    s += v.x * va.x + v.y * va.y + v.z * va.z + v.w * va.w;
    d += v.x * vb.x + v.y * vb.y + v.z * vb.z + v.w * vb.w;
  }
  as[i] = s;
  ad[i] = d;
}

// float4 fill (n4 = element count / 4)
__global__ void gat455_fill4(float4* __restrict__ p, int n4, float v) {
  int i = blockIdx.x * blockDim.x + threadIdx.x;
  if (i < n4) p[i] = make_float4(v, v, v, v);
}

__device__ __forceinline__ float gat455_leaky(float x) {
  return x > 0.f ? x : NEG_SLOPE * x;
}

// float atomic max via int/uint punning (valid with -inf init)
__device__ __forceinline__ void gat455_atomic_max_f32(float* addr, float val) {
  if (val >= 0.f)
    atomicMax((int*)addr, __float_as_int(val));
  else
    atomicMin((unsigned int*)addr, __float_as_uint(val));
}

// pass 1: segment max of leaky-relu logits over destination nodes
__global__ void gat455_edge_max(const int* __restrict__ srcp,
                                const int* __restrict__ dstp,
                                int E, int Etot,
                                const float* __restrict__ as,
                                const float* __restrict__ ad,
                                float* __restrict__ mv, int H) {
  int i = blockIdx.x * blockDim.x + threadIdx.x;
  if (i >= Etot * H) return;
  int e = i / H, hh = i - e * H;
  int s = (e < E) ? srcp[e] : (e - E);
  int d = (e < E) ? dstp[e] : (e - E);
  float logit = gat455_leaky(as[s * H + hh] + ad[d * H + hh]);
  gat455_atomic_max_f32(&mv[d * H + hh], logit);
}

// pass 2: segment sum of exp(logit - max)
__global__ void gat455_edge_denom(const int* __restrict__ srcp,
                                  const int* __restrict__ dstp,
                                  int E, int Etot,
                                  const float* __restrict__ as,
                                  const float* __restrict__ ad,
                                  const float* __restrict__ mv,
                                  float* __restrict__ dn, int H) {
  int i = blockIdx.x * blockDim.x + threadIdx.x;
  if (i >= Etot * H) return;
  int e = i / H, hh = i - e * H;
  int s = (e < E) ? srcp[e] : (e - E);
  int d = (e < E) ? dstp[e] : (e - E);
  float logit = gat455_leaky(as[s * H + hh] + ad[d * H + hh]);
  float p = __expf(logit - mv[d * H + hh]);
  atomicAdd(&dn[d * H + hh], p);
}

// ---------------------------------------------------------------------------
// pass 3: EPW edges per wave; 32/EPW lanes per edge; each lane owns 4
// consecutive channels -> one global_load_b128 gather of h[src] per lane and
// fully-coalesced 512B row transactions; 4 f32 atomics into L2-resident agg.
//   EPW=1 -> F=128 (layers 1/2),  EPW=4 -> F=32 (layer 3)
// ---------------------------------------------------------------------------
template <int EPW, int H, int Cc>
__global__ void gat455_edge_aggr(const int* __restrict__ srcp,
                                 const int* __restrict__ dstp,
                                 int E, int Etot,
                                 const float* __restrict__ as,
                                 const float* __restrict__ ad,
                                 const float* __restrict__ mv,
                                 const float* __restrict__ dn,
                                 const float* __restrict__ hproj,
                                 float* __restrict__ agg) {
  constexpr int LPE = 32 / EPW;   // lanes per edge
  constexpr int F   = LPE * 4;    // channels per edge
  int wave = (blockIdx.x * blockDim.x + threadIdx.x) >> 5;
  int lane = threadIdx.x & 31;
  int e = wave * EPW + lane / LPE;
  if (e >= Etot) return;
  int sub = lane % LPE;
  int c0  = sub << 2;
  int hh  = c0 / Cc;
  int s = (e < E) ? srcp[e] : (e - E);
  int d = (e < E) ? dstp[e] : (e - E);

  float logit = gat455_leaky(as[s * H + hh] + ad[d * H + hh]);
  float alpha = __expf(logit - mv[d * H + hh]) / (dn[d * H + hh] + 1e-16f);

  float4 hv = *(const float4*)(hproj + (size_t)s * F + c0);   // b128 gather
  float* ap = agg + (size_t)d * F + c0;
  atomicAdd(ap + 0, hv.x * alpha);
  atomicAdd(ap + 1, hv.y * alpha);
  atomicAdd(ap + 2, hv.z * alpha);
  atomicAdd(ap + 3, hv.w * alpha);
}

// out = agg + bias, optional ELU (float4 streams)
__global__ void gat455_finalize(const float4* __restrict__ agg,
                                const float* __restrict__ b,
                                float4* __restrict__ out,
                                int n4, int F4, int doElu) {
  int i = blockIdx.x * blockDim.x + threadIdx.x;
  if (i >= n4) return;
  int c = (i % F4) << 2;
  float4 v = agg[i];
  v.x += b[c + 0];
  v.y += b[c + 1];
  v.z += b[c + 2];
  v.w += b[c + 3];
  if (doElu) {
    v.x = v.x > 0.f ? v.x : (__expf(v.x) - 1.f);
    v.y = v.y > 0.f ? v.y : (__expf(v.y) - 1.f);
    v.z = v.z > 0.f ? v.z : (__expf(v.z) - 1.f);
    v.w = v.w > 0.f ? v.w : (__expf(v.w) - 1.f);
  }
  out[i] = v;
}

// ---------------------------------------------------------------------------
extern "C" void kernel_launch(void* const* d_in, const int* in_sizes, int n_in,
                              void* d_out, int out_size, void* d_ws, size_t ws_size,
                              hipStream_t stream) {
  const float* x    = (const float*)d_in[0];
  const int*   ei   = (const int*)d_in[1];
  const float* W1   = (const float*)d_in[2];
  const float* aS1  = (const float*)d_in[3];
  const float* aD1  = (const float*)d_in[4];
  const float* b1   = (const float*)d_in[5];
  const float* W2   = (const float*)d_in[6];
  const float* aS2  = (const float*)d_in[7];
  const float* aD2  = (const float*)d_in[8];
  const float* b2   = (const float*)d_in[9];
  const float* W3   = (const float*)d_in[10];
  const float* aS3  = (const float*)d_in[11];
  const float* aD3  = (const float*)d_in[12];
  const float* b3   = (const float*)d_in[13];

  const int IN = 128, HIDF = 128, H = 4, Cc = 32, OUTF = 32;
  const int NN = in_sizes[0] / IN;      // 100000
  const int E = in_sizes[1] / 2;        // 1600000
  const int Etot = E + NN;              // edges + self loops
  const int* srcp = ei;
  const int* dstp = ei + E;

  // workspace carve-up (~122 MB)
  char* ws = (char*)d_ws;
  size_t szBig   = (size_t)NN * HIDF * sizeof(float);  // 51.2 MB
  size_t szSmall = (size_t)NN * OUTF * sizeof(float);  // 12.8 MB
  size_t szAl    = (size_t)NN * H * sizeof(float);     // 1.6 MB
  float* B1 = (float*)ws;                              // features/agg ping-pong
  float* B2 = (float*)(ws + szBig);
  float* B3 = (float*)(ws + 2 * szBig);                // layer-3 projection
  float* asP = (float*)(ws + 2 * szBig + szSmall);
  float* adP = (float*)(ws + 2 * szBig + szSmall + szAl);
  float* mvP = (float*)(ws + 2 * szBig + szSmall + 2 * szAl);
  float* dnP = (float*)(ws + 2 * szBig + szSmall + 3 * szAl);

  const int TPB = 256;
  const int gemmBlocks = ((NN / 16) + 7) / 8;          // 8 waves/block
  auto cdiv = [](int a, int b) { return (a + b - 1) / b; };
  const int aggrBlocks128 = cdiv(Etot * 32, TPB);      // 1 edge / wave
  const int aggrBlocks32  = cdiv(cdiv(Etot, 4) * 32, TPB); // 4 edges / wave

  // ---------------- layer 1 : x -> B1(proj), agg in B2, out -> B1 ----------
  gat455_gemm_wmma<8><<<gemmBlocks, TPB, 0, stream>>>(x, W1, B1, NN, IN, HIDF);
  gat455_alpha<<<cdiv(NN * H, TPB), TPB, 0, stream>>>(B1, aS1, aD1, asP, adP, NN, H, Cc);
  gat455_fill4<<<cdiv(NN * H / 4, TPB), TPB, 0, stream>>>((float4*)mvP, NN * H / 4, -INFINITY);
  gat455_fill4<<<cdiv(NN * H / 4, TPB), TPB, 0, stream>>>((float4*)dnP, NN * H / 4, 0.f);
  gat455_fill4<<<cdiv(NN * HIDF / 4, TPB), TPB, 0, stream>>>((float4*)B2, NN * HIDF / 4, 0.f);
  gat455_edge_max<<<cdiv(Etot * H, TPB), TPB, 0, stream>>>(srcp, dstp, E, Etot, asP, adP, mvP, H);
  gat455_edge_denom<<<cdiv(Etot * H, TPB), TPB, 0, stream>>>(srcp, dstp, E, Etot, asP, adP, mvP, dnP, H);
  gat455_edge_aggr<1, 4, 32><<<aggrBlocks128, TPB, 0, stream>>>(srcp, dstp, E, Etot, asP, adP, mvP, dnP, B1, B2);
  gat455_finalize<<<cdiv(NN * HIDF / 4, TPB), TPB, 0, stream>>>((float4*)B2, b1, (float4*)B1,
                                                                NN * HIDF / 4, HIDF / 4, 1);

  // ---------------- layer 2 : B1 -> B2(proj), agg in B1, out -> B2 ---------
  gat455_gemm_wmma<8><<<gemmBlocks, TPB, 0, stream>>>(B1, W2, B2, NN, HIDF, HIDF);
  gat455_alpha<<<cdiv(NN * H, TPB), TPB, 0, stream>>>(B2, aS2, aD2, asP, adP, NN, H, Cc);
  gat455_fill4<<<cdiv(NN * H / 4, TPB), TPB, 0, stream>>>((float4*)mvP, NN * H / 4, -INFINITY);
  gat455_fill4<<<cdiv(NN * H / 4, TPB), TPB, 0, stream>>>((float4*)dnP, NN * H / 4, 0.f);
  gat455_fill4<<<cdiv(NN * HIDF / 4, TPB), TPB, 0, stream>>>((float4*)B1, NN * HIDF / 4, 0.f);
  gat455_edge_max<<<cdiv(Etot * H, TPB), TPB, 0, stream>>>(srcp, dstp, E, Etot, asP, adP, mvP, H);
  gat455_edge_denom<<<cdiv(Etot * H, TPB), TPB, 0, stream>>>(srcp, dstp, E, Etot, asP, adP, mvP, dnP, H);
  gat455_edge_aggr<1, 4, 32><<<aggrBlocks128, TPB, 0, stream>>>(srcp, dstp, E, Etot, asP, adP, mvP, dnP, B2, B1);
  gat455_finalize<<<cdiv(NN * HIDF / 4, TPB), TPB, 0, stream>>>((float4*)B1, b2, (float4*)B2,
                                                                NN * HIDF / 4, HIDF / 4, 1);

  // ---------------- layer 3 : B2 -> B3(proj), agg in B1, out -> d_out ------
  gat455_gemm_wmma<2><<<gemmBlocks, TPB, 0, stream>>>(B2, W3, B3, NN, HIDF, OUTF);
  gat455_alpha<<<cdiv(NN, TPB), TPB, 0, stream>>>(B3, aS3, aD3, asP, adP, NN, 1, OUTF);
  gat455_fill4<<<cdiv(NN / 4, TPB), TPB, 0, stream>>>((float4*)mvP, NN / 4, -INFINITY);
  gat455_fill4<<<cdiv(NN / 4, TPB), TPB, 0, stream>>>((float4*)dnP, NN / 4, 0.f);
  gat455_fill4<<<cdiv(NN * OUTF / 4, TPB), TPB, 0, stream>>>((float4*)B1, NN * OUTF / 4, 0.f);
  gat455_edge_max<<<cdiv(Etot, TPB), TPB, 0, stream>>>(srcp, dstp, E, Etot, asP, adP, mvP, 1);
  gat455_edge_denom<<<cdiv(Etot, TPB), TPB, 0, stream>>>(srcp, dstp, E, Etot, asP, adP, mvP, dnP, 1);
  gat455_edge_aggr<4, 1, 32><<<aggrBlocks32, TPB, 0, stream>>>(srcp, dstp, E, Etot, asP, adP, mvP, dnP, B3, B1);
  gat455_finalize<<<cdiv(NN * OUTF / 4, TPB), TPB, 0, stream>>>((float4*)B1, b3, (float4*)d_out,
                                                                NN * OUTF / 4, OUTF / 4, 0);
}